// GCN_25890062861000
// MI455X (gfx1250) — compile-verified
//
#include <hip/hip_runtime.h>

// Problem constants (fixed by the reference).
#define B_ 4
#define N_ 50000
#define E_ 1600000
#define FIN_ 12
#define FOUT_ 48
#define SEQ_ 12
#define DHEAD_ 4   // FOUT_ / SEQ_

typedef __attribute__((ext_vector_type(2))) float v2f;
typedef __attribute__((ext_vector_type(8))) float v8f;

// Hardware f32 atomic add (global_atomic_add_f32), agent scope, relaxed.
__device__ __forceinline__ void atomicAddF32(float* p, float v) {
    __hip_atomic_fetch_add(p, v, __ATOMIC_RELAXED, __HIP_MEMORY_SCOPE_AGENT);
}

// ---------------------------------------------------------------------------
// 1) deg[b*N+n] = 1.0f  (self-loop weight)
__global__ __launch_bounds__(256) void k_init_deg(float* deg) {
    int i = blockIdx.x * blockDim.x + threadIdx.x;
    if (i < B_ * N_) deg[i] = 1.0f;
}

// 2) deg[b*N + dst] += ew   (one thread per edge)
__global__ __launch_bounds__(256) void k_deg_scatter(const int* __restrict__ ei,
                                                     const float* __restrict__ ea,
                                                     float* __restrict__ deg) {
    int i = blockIdx.x * blockDim.x + threadIdx.x;
    if (i >= B_ * E_) return;
    int b = i / E_;
    int e = i - b * E_;
    const int* eib = ei + b * 2 * E_;
    int dst = eib[E_ + e];
    atomicAddF32(&deg[b * N_ + dst], ea[(size_t)b * E_ + e]);
}

// 3) dinv = rsqrt(deg)
__global__ __launch_bounds__(256) void k_dinv(const float* __restrict__ deg,
                                              float* __restrict__ dinv) {
    int i = blockIdx.x * blockDim.x + threadIdx.x;
    if (i < B_ * N_) dinv[i] = __frsqrt_rn(deg[i]);
}

// ---------------------------------------------------------------------------
// 4) xw = x @ W via V_WMMA_F32_16X16X4_F32.
//    One wave per 16-node x 16-feature tile; K=12 done as 3 chained K=4 WMMAs.
//    N_=3125*16 and FOUT_=3*16 tile exactly (no predication; EXEC all ones).
__global__ __launch_bounds__(256) void k_xw_wmma(const float* __restrict__ x,
                                                 const float* __restrict__ W,
                                                 float* __restrict__ xw) {
    const int lane = threadIdx.x & 31;
    const int wave = (blockIdx.x * blockDim.x + threadIdx.x) >> 5;
    const int tiles_per_b = (N_ / 16) * (FOUT_ / 16);   // 3125 * 3
    const int b = wave / tiles_per_b;
    if (b >= B_) return;                                 // wave-uniform exit
    const int t        = wave - b * tiles_per_b;
    const int row_tile = t / (FOUT_ / 16);
    const int col_tile = t - row_tile * (FOUT_ / 16);
    const int node0 = row_tile * 16;
    const int col0  = col_tile * 16;
    const int hl = lane & 15;     // M (for A) / N (for B,C,D) index
    const int hi = lane >> 4;     // selects K pair 0/1 vs 2/3 ; row offset +8 for C/D

    const float* xrow = x + ((size_t)b * N_ + node0 + hl) * FIN_;
    v8f acc = {};
#pragma unroll
    for (int kb = 0; kb < FIN_; kb += 4) {
        // A: 16x4 f32 — lanes 0-15: K=kb,kb+1 ; lanes 16-31: K=kb+2,kb+3
        v2f a;
        a.x = xrow[kb + 2 * hi];
        a.y = xrow[kb + 2 * hi + 1];
        // B: 4x16 f32 — mirrored layout, N across lanes 0-15, K split by half-wave
        v2f bm;
        bm.x = W[(kb + 2 * hi) * FOUT_ + col0 + hl];
        bm.y = W[(kb + 2 * hi + 1) * FOUT_ + col0 + hl];
        acc = __builtin_amdgcn_wmma_f32_16x16x4_f32(
            /*neg_a=*/false, a, /*neg_b=*/false, bm,
            /*c_mod=*/(short)0, acc, /*reuse_a=*/false, /*reuse_b=*/false);
    }
    // D layout: VGPR r, lanes 0-15 -> row r, lanes 16-31 -> row r+8, col = hl
    float* orow = xw + ((size_t)b * N_ + node0) * FOUT_ + col0;
#pragma unroll
    for (int r = 0; r < 8; ++r) {
        orow[(size_t)(r + 8 * hi) * FOUT_ + hl] = acc[r];
    }
}

// ---------------------------------------------------------------------------
// 5) Self-loop + bias, written directly in the permuted output layout
//    out[b][s][n][d] = xw[b][n][s*4+d] * dinv^2 + bias[s*4+d]
//    Thread id == linear out index -> perfectly coalesced stores.
__global__ __launch_bounds__(256) void k_self_out(const float* __restrict__ xw,
                                                  const float* __restrict__ dinv,
                                                  const float* __restrict__ bias,
                                                  float* __restrict__ out) {
    int i = blockIdx.x * blockDim.x + threadIdx.x;   // over B*SEQ*N*DHEAD = 9.6M
    if (i >= B_ * SEQ_ * N_ * DHEAD_) return;
    int d  = i & (DHEAD_ - 1);
    int t  = i >> 2;            // DHEAD_ == 4
    int n  = t % N_;
    int t2 = t / N_;
    int s  = t2 % SEQ_;
    int b  = t2 / SEQ_;
    int f  = s * DHEAD_ + d;
    float di = dinv[b * N_ + n];
    out[i] = xw[((size_t)b * N_ + n) * FOUT_ + f] * (di * di) + bias[f];
}

// ---------------------------------------------------------------------------
// 6) Edge messages: one edge per 16-lane half-wave (2 edges / wave).
//    Each lane handles 3 of the 48 features -> gathers of xw[src] are three
//    64B segments; atomic scatter-adds land in 16B segments of the permuted
//    output. Everything stays L2-resident (<=40MB working set vs 192MB L2).
__global__ __launch_bounds__(256) void k_edge_msg(const int* __restrict__ ei,
                                                  const float* __restrict__ ea,
                                                  const float* __restrict__ xw,
                                                  const float* __restrict__ dinv,
                                                  float* __restrict__ out) {
    int half = (blockIdx.x * blockDim.x + threadIdx.x) >> 4;  // edge slot
    if (half >= B_ * E_) return;
    int b = half / E_;
    int e = half - b * E_;
    int hl = threadIdx.x & 15;

    const int* eib = ei + b * 2 * E_;
    int src = eib[e];
    int dst = eib[E_ + e];
    float w = ea[(size_t)b * E_ + e];
    float norm = dinv[b * N_ + src] * w * dinv[b * N_ + dst];

    const float* xr = xw + ((size_t)b * N_ + src) * FOUT_;
#pragma unroll
    for (int j = 0; j < 3; ++j) {
        int f = hl + 16 * j;
        float v = xr[f] * norm;
        int s = f >> 2;           // seq slot
        int d = f & 3;            // head dim
        float* o = out + (((size_t)b * SEQ_ + s) * N_ + dst) * DHEAD_ + d;
        atomicAddF32(o, v);
    }
}

// ---------------------------------------------------------------------------
extern "C" void kernel_launch(void* const* d_in, const int* in_sizes, int n_in,
                              void* d_out, int out_size, void* d_ws, size_t ws_size,
                              hipStream_t stream) {
    const float* x    = (const float*)d_in[0];   // [B,N,12]
    const int*   ei   = (const int*)  d_in[1];   // [B,2,E]
    const float* ea   = (const float*)d_in[2];   // [B,E]
    const float* W    = (const float*)d_in[3];   // [12,48]
    const float* bias = (const float*)d_in[4];   // [48]
    float* out = (float*)d_out;                  // [B,12,N,4]

    // Workspace layout (fp32): xw [B*N*48] | deg [B*N] | dinv [B*N]  (~40MB)
    float* xw   = (float*)d_ws;
    float* deg  = xw  + (size_t)B_ * N_ * FOUT_;
    float* dinv = deg + (size_t)B_ * N_;

    const int TB = 256;
    // 1) deg = 1
    k_init_deg<<<(B_ * N_ + TB - 1) / TB, TB, 0, stream>>>(deg);
    // 2) deg += ew @ dst
    k_deg_scatter<<<(B_ * E_ + TB - 1) / TB, TB, 0, stream>>>(ei, ea, deg);
    // 3) dinv = rsqrt(deg)
    k_dinv<<<(B_ * N_ + TB - 1) / TB, TB, 0, stream>>>(deg, dinv);
    // 4) xw = x @ W  (WMMA f32 16x16x4) — 37500 waves, 8 waves/block
    {
        int waves = B_ * (N_ / 16) * (FOUT_ / 16);
        int blocks = (waves + (TB / 32) - 1) / (TB / 32);
        k_xw_wmma<<<blocks, TB, 0, stream>>>(x, W, xw);
    }
    // 5) out = selfloop + bias (permuted layout, full overwrite)
    k_self_out<<<(B_ * SEQ_ * N_ * DHEAD_ + TB - 1) / TB, TB, 0, stream>>>(xw, dinv, bias, out);
    // 6) atomic edge scatter into out
    {
        long long threads = (long long)B_ * E_ * 16;   // 16 lanes per edge
        int blocks = (int)((threads + TB - 1) / TB);
        k_edge_msg<<<blocks, TB, 0, stream>>>(ei, ea, xw, dinv, out);
    }
}